// AdditiveAttention_55473797595831
// MI455X (gfx1250) — compile-verified
//
#include <hip/hip_runtime.h>

// ---------------------------------------------------------------------------
// AdditiveAttention for MI455X (gfx1250, wave32)
//   B=4, Q=256, K=1024, DQ=DK=DV=512, H=256
//
//   Phase 1: qproj[b,q,h]  = sum_d queries[b,q,d] * Wq[h,d]   (WMMA f32 16x16x4)
//            kprojT[b,h,k] = sum_d keys[b,k,d]    * Wk[h,d]   (WMMA, transposed store)
//   Phase 2: scores[b,q,k] = sum_h tanh(qproj+kprojT)*wv[h], mask, softmax -> attn
//   Phase 3: out[b,q,v]    = sum_k attn[b,q,k] * values[b,k,v] (WMMA f32 16x16x4)
// ---------------------------------------------------------------------------

typedef float v2f __attribute__((ext_vector_type(2)));
typedef float v8f __attribute__((ext_vector_type(8)));

#define BATCH 4
#define QLEN  256
#define KLEN  1024
#define DQK   512
#define DV    512
#define HID   256

#if __has_builtin(__builtin_amdgcn_tanhf)
__device__ __forceinline__ float fast_tanh(float x) { return __builtin_amdgcn_tanhf(x); }
#else
__device__ __forceinline__ float fast_tanh(float x) { return tanhf(x); }
#endif

// ---------------------------------------------------------------------------
// C[b] = A[b] (MxK, row major) * W^T   with W (NxK, row major).
// transposeStore==0 : C layout [B, M, N]
// transposeStore==1 : C layout [B, N, M]
// One wave32 per 16x16 output tile. Exact grid: gridDim.x * 4 waves == tiles.
// ---------------------------------------------------------------------------
__global__ void gemm_awt_wmma(const float* __restrict__ A,
                              const float* __restrict__ W,
                              float* __restrict__ C,
                              int M, int N, int Kd, int transposeStore) {
    const int lane = threadIdx.x & 31;
    const int wave = threadIdx.x >> 5;
    const int tilesN = N >> 4;
    const int tilesPerBatch = (M >> 4) * tilesN;

    int tid = blockIdx.x * (blockDim.x >> 5) + wave;
    const int b  = tid / tilesPerBatch;
    const int r  = tid % tilesPerBatch;
    const int tm = r / tilesN;
    const int tn = r % tilesN;

    const float* Ab = A + (size_t)b * M * Kd;
    const int row  = (tm << 4) + (lane & 15);          // A row (M index)
    const int col  = (tn << 4) + (lane & 15);          // W row (N index)
    const int kph  = (lane >> 4) << 1;                 // K phase: 0 or 2

    const float* arow = Ab + (size_t)row * Kd + kph;
    const float* wrow = W  + (size_t)col * Kd + kph;

    v8f c = {0.f, 0.f, 0.f, 0.f, 0.f, 0.f, 0.f, 0.f};
    for (int k = 0; k < Kd; k += 4) {
        v2f a  = *(const v2f*)(arow + k);   // A[row][k+kph], A[row][k+kph+1]
        v2f bb = *(const v2f*)(wrow + k);   // W[col][k+kph], W[col][k+kph+1] == B^T
        c = __builtin_amdgcn_wmma_f32_16x16x4_f32(false, a, false, bb,
                                                  (short)0, c, false, false);
    }

    // D layout: VGPR j -> lanes 0-15: (M=j, N=lane); lanes 16-31: (M=j+8, N=lane-16)
    const int mhi = (lane >> 4) << 3;       // +8 for upper half-wave
    float* Cb = C + (size_t)b * M * N;
    if (!transposeStore) {
#pragma unroll
        for (int j = 0; j < 8; ++j) {
            int mm = (tm << 4) + j + mhi;
            int nn = (tn << 4) + (lane & 15);
            Cb[(size_t)mm * N + nn] = c[j];
        }
    } else {
#pragma unroll
        for (int j = 0; j < 8; ++j) {
            int mm = (tm << 4) + j + mhi;
            int nn = (tn << 4) + (lane & 15);
            Cb[(size_t)nn * M + mm] = c[j];  // store [B, N, M]
        }
    }
}

// ---------------------------------------------------------------------------
// Fused score + mask + softmax.  One 256-thread block per (b, q) row.
// Each thread owns 4 key positions {t, t+256, t+512, t+768}.
// 268M tanh total across the grid -> TRANS pipe bound; kprojT reads are
// coalesced and L2 resident (4 MB).
// ---------------------------------------------------------------------------
__global__ void score_softmax_kernel(const float* __restrict__ qproj,   // [B,Q,H]
                                     const float* __restrict__ kprojT,  // [B,H,K]
                                     const float* __restrict__ wv,      // [H]
                                     const int*   __restrict__ valid_lens, // [B]
                                     float* __restrict__ attn) {        // [B,Q,K]
    const int bq = blockIdx.x;            // 0 .. B*Q-1
    const int b  = bq >> 8;               // Q == 256
    const int t  = threadIdx.x;           // 0 .. 255

    __shared__ float qs[HID];
    __shared__ float wvs[HID];
    __shared__ float red[256];

    qs[t]  = qproj[(size_t)bq * HID + t];
    wvs[t] = wv[t];
    __syncthreads();

    const float* kb = kprojT + (size_t)b * HID * KLEN;
    float s0 = 0.f, s1 = 0.f, s2 = 0.f, s3 = 0.f;
    for (int h = 0; h < HID; ++h) {
        const float qh = qs[h];
        const float w  = wvs[h];
        const float* row = kb + (size_t)h * KLEN;
        s0 = fmaf(fast_tanh(qh + row[t       ]), w, s0);
        s1 = fmaf(fast_tanh(qh + row[t + 256 ]), w, s1);
        s2 = fmaf(fast_tanh(qh + row[t + 512 ]), w, s2);
        s3 = fmaf(fast_tanh(qh + row[t + 768 ]), w, s3);
    }

    const int vl = valid_lens[b];
    if (t        >= vl) s0 = -1e6f;
    if (t + 256  >= vl) s1 = -1e6f;
    if (t + 512  >= vl) s2 = -1e6f;
    if (t + 768  >= vl) s3 = -1e6f;

    // --- block max ---
    float m = fmaxf(fmaxf(s0, s1), fmaxf(s2, s3));
    red[t] = m;
    __syncthreads();
    for (int off = 128; off > 0; off >>= 1) {
        if (t < off) red[t] = fmaxf(red[t], red[t + off]);
        __syncthreads();
    }
    m = red[0];
    __syncthreads();

    // --- exp + block sum ---
    const float e0 = __expf(s0 - m);
    const float e1 = __expf(s1 - m);
    const float e2 = __expf(s2 - m);
    const float e3 = __expf(s3 - m);
    red[t] = e0 + e1 + e2 + e3;
    __syncthreads();
    for (int off = 128; off > 0; off >>= 1) {
        if (t < off) red[t] += red[t + off];
        __syncthreads();
    }
    const float inv = 1.0f / red[0];

    float* arow = attn + (size_t)bq * KLEN;
    arow[t       ] = e0 * inv;
    arow[t + 256 ] = e1 * inv;
    arow[t + 512 ] = e2 * inv;
    arow[t + 768 ] = e3 * inv;
}

// ---------------------------------------------------------------------------
// out[b] = attn[b] (MxK) * V[b] (KxN).  One wave32 per 16x16 tile.
// ---------------------------------------------------------------------------
__global__ void gemm_av_wmma(const float* __restrict__ Amat,  // [B,M,K]
                             const float* __restrict__ V,     // [B,K,N]
                             float* __restrict__ C,           // [B,M,N]
                             int M, int N, int Kd) {
    const int lane = threadIdx.x & 31;
    const int wave = threadIdx.x >> 5;
    const int tilesN = N >> 4;
    const int tilesPerBatch = (M >> 4) * tilesN;

    int tid = blockIdx.x * (blockDim.x >> 5) + wave;
    const int b  = tid / tilesPerBatch;
    const int r  = tid % tilesPerBatch;
    const int tm = r / tilesN;
    const int tn = r % tilesN;

    const float* Ab = Amat + (size_t)b * M * Kd;
    const float* Vb = V    + (size_t)b * Kd * N;
    const int row = (tm << 4) + (lane & 15);
    const int col = (tn << 4) + (lane & 15);
    const int kph = (lane >> 4) << 1;                  // 0 or 2

    const float* arow = Ab + (size_t)row * Kd + kph;
    const float* vcol = Vb + (size_t)kph * N + col;

    v8f c = {0.f, 0.f, 0.f, 0.f, 0.f, 0.f, 0.f, 0.f};
    for (int k = 0; k < Kd; k += 4) {
        v2f a = *(const v2f*)(arow + k);               // attn[row][k+kph .. +1]
        v2f bb;
        bb.x = vcol[(size_t)k * N];                    // V[k+kph  ][col]
        bb.y = vcol[(size_t)k * N + N];                // V[k+kph+1][col]
        c = __builtin_amdgcn_wmma_f32_16x16x4_f32(false, a, false, bb,
                                                  (short)0, c, false, false);
    }

    const int mhi = (lane >> 4) << 3;
    float* Cb = C + (size_t)b * M * N;
#pragma unroll
    for (int j = 0; j < 8; ++j) {
        int mm = (tm << 4) + j + mhi;
        int nn = (tn << 4) + (lane & 15);
        Cb[(size_t)mm * N + nn] = c[j];
    }
}

// ---------------------------------------------------------------------------
extern "C" void kernel_launch(void* const* d_in, const int* in_sizes, int n_in,
                              void* d_out, int out_size, void* d_ws, size_t ws_size,
                              hipStream_t stream) {
    const float* queries    = (const float*)d_in[0];  // [B,Q,DQ]
    const float* keys       = (const float*)d_in[1];  // [B,K,DK]
    const float* values     = (const float*)d_in[2];  // [B,K,DV]
    const int*   valid_lens = (const int*)  d_in[3];  // [B]
    const float* Wq         = (const float*)d_in[4];  // [H,DQ]
    const float* Wk         = (const float*)d_in[5];  // [H,DK]
    const float* wv         = (const float*)d_in[6];  // [H]
    float* out = (float*)d_out;                       // [B,Q,DV]

    float* ws     = (float*)d_ws;
    float* qproj  = ws;                               // B*Q*H  = 262144 f32
    float* kprojT = qproj  + (size_t)BATCH * QLEN * HID;   // B*H*K = 1048576 f32
    float* attn   = kprojT + (size_t)BATCH * HID * KLEN;   // B*Q*K = 1048576 f32

    // qproj: tiles = 4 * (256/16) * (256/16) = 1024 waves -> 256 blocks x 4 waves
    gemm_awt_wmma<<<256, 128, 0, stream>>>(queries, Wq, qproj,
                                           QLEN, HID, DQK, 0);
    // kprojT: tiles = 4 * (1024/16) * (256/16) = 4096 waves -> 1024 blocks
    gemm_awt_wmma<<<1024, 128, 0, stream>>>(keys, Wk, kprojT,
                                            KLEN, HID, DQK, 1);
    // one block per (b,q) row: B*Q = 1024 blocks of 256 threads
    score_softmax_kernel<<<BATCH * QLEN, 256, 0, stream>>>(qproj, kprojT, wv,
                                                           valid_lens, attn);
    // out: tiles = 4 * (256/16) * (512/16) = 2048 waves -> 512 blocks
    gemm_av_wmma<<<512, 128, 0, stream>>>(attn, values, out,
                                          QLEN, DV, KLEN);
}